// LocSE_32607391711324
// MI455X (gfx1250) — compile-verified
//
#include <hip/hip_runtime.h>

// LocSE for MI455X (gfx1250, wave32).
// Kernel 1: per-thread KNN top-16 scan with full batch coords staged in LDS.
// Kernel 2: one wave per point; relative-position encoding -> 1x1 conv via
//           V_WMMA_F32_16X16X4_F32 (A = W tile 16x4, B = enc^T 4x16), bias
//           folded into the accumulator init; feature concat broadcast.
// Workspace use: B*N*K ints + B*N*K floats = 2 MB.

typedef __attribute__((ext_vector_type(2))) float v2f;
typedef __attribute__((ext_vector_type(8))) float v8f;

#define B_SZ   4
#define N_PTS  4096
#define D_FEAT 64
#define D_OUT  128
#define KNN    16

// ---------------------------------------------------------------------------
// Kernel 1: KNN (one thread per query point, one batch per blockIdx.y)
// ---------------------------------------------------------------------------
__global__ __launch_bounds__(256) void knn_kernel(const float* __restrict__ coords,
                                                  int* __restrict__ knn_idx,
                                                  float* __restrict__ knn_dist) {
    __shared__ float sx[N_PTS], sy[N_PTS], sz[N_PTS], sw[N_PTS];  // 64 KB
    const int tid = threadIdx.x;
    const int b   = blockIdx.y;
    const float* cb = coords + (size_t)b * N_PTS * 3;

    for (int p = tid; p < N_PTS; p += 256) {
        float x = cb[p * 3 + 0];
        float y = cb[p * 3 + 1];
        float z = cb[p * 3 + 2];
        sx[p] = x; sy[p] = y; sz[p] = z;
        sw[p] = fmaf(z, z, fmaf(y, y, x * x));
    }
    __syncthreads();

    const int i = blockIdx.x * 256 + tid;           // query point
    const float qx = sx[i], qy = sy[i], qz = sz[i], qsq = sw[i];

    float bd[KNN];
    int   bi[KNN];
#pragma unroll
    for (int s = 0; s < KNN; ++s) { bd[s] = 3.0e38f; bi[s] = 0; }

    for (int jp = 0; jp < N_PTS; ++jp) {
        // reference formula: |i|^2 + |j|^2 - 2 dot  (self distance == 0 exactly)
        float dot = fmaf(sz[jp], qz, fmaf(sy[jp], qy, sx[jp] * qx));
        float d2  = fmaxf(qsq + sw[jp] - 2.0f * dot, 0.0f);
        if (d2 < bd[KNN - 1]) {
            float dc = d2; int ic = jp;
#pragma unroll
            for (int s = 0; s < KNN; ++s) {
                bool  sw_ = dc < bd[s];            // strict < keeps top_k tie order
                float td  = sw_ ? bd[s] : dc;
                bd[s]     = sw_ ? dc    : bd[s];
                dc = td;
                int   ti  = sw_ ? bi[s] : ic;
                bi[s]     = sw_ ? ic    : bi[s];
                ic = ti;
            }
        }
    }

    const size_t base = ((size_t)b * N_PTS + i) * KNN;
#pragma unroll
    for (int s = 0; s < KNN; ++s) {
        knn_idx[base + s]  = bi[s];
        knn_dist[base + s] = sqrtf(fmaxf(bd[s], 1e-12f));
    }
}

// ---------------------------------------------------------------------------
// Kernel 2: encoding + 1x1 conv via f32 WMMA + feature concat
//   A (16x4):  lane holds row M = lane&15,  vgpr v holds K = v + 2*(lane>>4)
//   B (4x16):  lane holds col N = lane&15,  vgpr v holds K = v + 2*(lane>>4)
//   D (16x16): lane holds col N = lane&15,  vgpr r holds M = r + 8*(lane>>4)
// ---------------------------------------------------------------------------
__global__ __launch_bounds__(256) void locse_mlp_kernel(const float* __restrict__ coords,
                                                        const float* __restrict__ features,
                                                        const float* __restrict__ Wm,
                                                        const float* __restrict__ bias,
                                                        const int* __restrict__ knn_idx,
                                                        const float* __restrict__ knn_dist,
                                                        float* __restrict__ out) {
    const int gtid = blockIdx.x * 256 + threadIdx.x;
    const int wave = gtid >> 5;                 // one wave per (b, n)
    const int lane = threadIdx.x & 31;
    const int b    = wave >> 12;                // / 4096
    const int n    = wave & (N_PTS - 1);
    const int j    = lane & 15;                 // neighbor slot == matrix N
    const int half = lane >> 4;
    const bool hi  = (half != 0);

    const size_t pbase = (size_t)b * N_PTS + n;
    const int   nidx = knn_idx[pbase * KNN + j];
    const float nd   = knn_dist[pbase * KNN + j];

    const float* qc = coords + pbase * 3;
    const float qx = qc[0], qy = qc[1], qz = qc[2];
    const float* pc = coords + ((size_t)b * N_PTS + nidx) * 3;
    const float px = pc[0], py = pc[1], pz = pc[2];

    // 10 encoding channels (+2 zero pad): [p_i, p_j, p_i - p_j, dist]
    const float e0 = qx, e1 = qy, e2 = qz;
    const float e3 = px, e4 = py, e5 = pz;
    const float e6 = qx - px, e7 = qy - py, e8 = qz - pz;
    const float e9 = nd;

    // B-matrix chunks: chunk c covers channels c*4 + (v + 2*half)
    v2f bv0, bv1, bv2;
    bv0[0] = hi ? e2  : e0;  bv0[1] = hi ? e3  : e1;
    bv1[0] = hi ? e6  : e4;  bv1[1] = hi ? e7  : e5;
    bv2[0] = hi ? 0.f : e8;  bv2[1] = hi ? 0.f : e9;

    const int chb = half * 2;

#pragma unroll
    for (int t = 0; t < 8; ++t) {               // 8 tiles of 16 output channels
        const int wrow = t * 16 + j;            // A row M = lane&15
        // Load all needed W entries unconditionally (always in-bounds: index
        // <= 127*10+9), then zero-pad channels 10/11 with plain cndmask —
        // avoids predicated loads / exec-mask branches between the WMMAs.
        const float w8 = Wm[wrow * 10 + 8];
        const float w9 = Wm[wrow * 10 + 9];
        v2f a0, a1, a2;
        a0[0] = Wm[wrow * 10 + chb];
        a0[1] = Wm[wrow * 10 + chb + 1];
        a1[0] = Wm[wrow * 10 + 4 + chb];
        a1[1] = Wm[wrow * 10 + 4 + chb + 1];
        a2[0] = hi ? 0.f : w8;                  // half=1 needs ch 10,11 -> 0
        a2[1] = hi ? 0.f : w9;

        v8f acc;                                 // seed with bias: C[M][N] = b[oc]
#pragma unroll
        for (int r = 0; r < 8; ++r) acc[r] = bias[t * 16 + half * 8 + r];

        acc = __builtin_amdgcn_wmma_f32_16x16x4_f32(false, a0, false, bv0, (short)0, acc, false, false);
        acc = __builtin_amdgcn_wmma_f32_16x16x4_f32(false, a1, false, bv1, (short)0, acc, false, false);
        acc = __builtin_amdgcn_wmma_f32_16x16x4_f32(false, a2, false, bv2, (short)0, acc, false, false);

        // out[b, oc, n, k] with oc = t*16 + 8*half + r, k = j
        const size_t obase =
            (((size_t)b * (D_OUT + D_FEAT) + t * 16 + half * 8) * N_PTS + n) * KNN + j;
#pragma unroll
        for (int r = 0; r < 8; ++r)
            out[obase + (size_t)r * (N_PTS * KNN)] = acc[r];
    }

    // Feature concat: out[b, 128 + c, n, k] = features[b, c, n]
#pragma unroll 4
    for (int it = 0; it < 32; ++it) {
        const int c = half + 2 * it;            // halves split even/odd channels
        const float v = features[((size_t)b * D_FEAT + c) * N_PTS + n];
        out[(((size_t)b * (D_OUT + D_FEAT) + D_OUT + c) * N_PTS + n) * KNN + j] = v;
    }
}

// ---------------------------------------------------------------------------
extern "C" void kernel_launch(void* const* d_in, const int* in_sizes, int n_in,
                              void* d_out, int out_size, void* d_ws, size_t ws_size,
                              hipStream_t stream) {
    (void)in_sizes; (void)n_in; (void)out_size; (void)ws_size;
    const float* coords   = (const float*)d_in[0];   // [B, N, 3]
    const float* features = (const float*)d_in[1];   // [B, 64, N]
    const float* Wm       = (const float*)d_in[2];   // [128, 10]
    const float* bias     = (const float*)d_in[3];   // [128]
    float* out = (float*)d_out;                      // [B, 192, N, 16]

    int*   knn_idx  = (int*)d_ws;                                            // 1 MB
    float* knn_dist = (float*)((char*)d_ws +
                               (size_t)B_SZ * N_PTS * KNN * sizeof(int));    // 1 MB

    dim3 g1(N_PTS / 256, B_SZ);
    knn_kernel<<<g1, 256, 0, stream>>>(coords, knn_idx, knn_dist);

    const int waves  = B_SZ * N_PTS;     // one wave per point
    const int blocks = waves / 8;        // 8 waves per 256-thread block
    locse_mlp_kernel<<<blocks, 256, 0, stream>>>(coords, features, Wm, bias,
                                                 knn_idx, knn_dist, out);
}